// NonInvariantPointAttention_9079560864171
// MI455X (gfx1250) — compile-verified
//
#include <hip/hip_runtime.h>
#include <hip/hip_bf16.h>
#include <math.h>

// ---------------- constants (shapes from reference) ----------------
#define LSEQ 768
#define DDIM 384
#define NH   12
#define DH   32
#define CZD  64
#define NPQ  4
#define NPV  8

typedef __attribute__((ext_vector_type(2))) float v2f;
typedef __attribute__((ext_vector_type(8))) float v8f;
typedef __attribute__((ext_vector_type(4))) unsigned int v4u;
typedef __attribute__((ext_vector_type(8))) int v8i;
typedef __attribute__((ext_vector_type(4))) int v4i;

__device__ __forceinline__ v8f wmma_f32(v2f a, v2f b, v8f c) {
  // V_WMMA_F32_16X16X4_F32 : D = A(16x4) * B(4x16) + C(16x16)
  return __builtin_amdgcn_wmma_f32_16x16x4_f32(
      /*neg_a=*/false, a, /*neg_b=*/false, b,
      /*c_mod=*/(short)0, c, /*reuse_a=*/false, /*reuse_b=*/false);
}

// ---------------- TDM 1-D tile load: global -> LDS ----------------
// Builds a D# descriptor (ISA cdna5 ch.8): count=1, type=2 ("image"),
// data_size=4B, 1-D tile of `nelem` dwords (tile_dim0=nelem, tile_dim1 unused).
// amdgpu-toolchain (clang-23) builtin arity: (v4u, v8i, v4i, v4i, v8i, i32 cpol).
__device__ __forceinline__ void tdm_load_1d(const void* gsrc, void* ldst, unsigned nelem) {
#if __has_builtin(__builtin_amdgcn_tensor_load_to_lds)
  typedef __attribute__((address_space(3))) char lds_char;
  unsigned lds_addr = (unsigned)(lds_char*)ldst;             // workgroup LDS byte offset
  unsigned long long ga = (unsigned long long)gsrc;          // global byte address

  v4u g0;
  g0.x = 1u;                                                  // count=1 (valid), no gather
  g0.y = lds_addr;                                            // lds_addr [63:32]
  g0.z = (unsigned)(ga & 0xffffffffu);                        // global_addr [95:64]
  g0.w = (unsigned)((ga >> 32) & 0x01ffffffu) | (2u << 30);   // global_addr hi + type=2

  v8i g1;
  g1[0] = (int)(2u << 16);                                    // wg_mask=0, data_size=2 (4B)
  g1[1] = (int)((nelem & 0xffffu) << 16);                     // tensor_dim0[15:0]
  g1[2] = (int)(((nelem >> 16) & 0xffffu) | (1u << 16));      // tensor_dim0[31:16], tensor_dim1=1
  g1[3] = (int)((nelem & 0xffffu) << 16);                     // tile_dim0 = nelem
  g1[4] = 0;                                                  // tile_dim1=0 (unused), tile_dim2=0
  g1[5] = (int)nelem;                                         // tensor_dim0_stride lo
  g1[6] = 0;
  g1[7] = 0;

  v4i g2 = {0, 0, 0, 0};
  v4i g3 = {0, 0, 0, 0};
  v8i g4 = {0, 0, 0, 0, 0, 0, 0, 0};
  __builtin_amdgcn_tensor_load_to_lds(g0, g1, g2, g3, g4, 0);
#else
  (void)gsrc; (void)ldst; (void)nelem;
#endif
}

// ---------------- generic batched WMMA GEMM (64x16 C per wave) ----------------
// C[M x N] (+)= alpha * A[M x K] * op(B) + bias
// TRANS_B=1: B is (N x K) row-major (weight layout, C = A * B^T)
// TRANS_B=0: B is (K x N) row-major
// Each wave: 4 M-subtiles x 1 N-tile; B fragment loaded once per K-step and
// reused by 4 independent v_wmma_f32_16x16x4_f32 (no RAW between them).
template<bool TRANS_B, bool ACCUM>
__global__ void gemm_wmma(const float* __restrict__ A, int lda, long long sA,
                          const float* __restrict__ B, int ldb, long long sB,
                          float* __restrict__ C, int ldc, long long sC,
                          const float* __restrict__ bias,
                          int M, int N, int K, float alpha)
{
  A += (long long)blockIdx.z * sA;
  B += (long long)blockIdx.z * sB;
  C += (long long)blockIdx.z * sC;

  int ntn = N >> 4;
  int tn  = blockIdx.x * (blockDim.x >> 5) + (threadIdx.x >> 5);
  if (tn >= ntn) return;
  int n0 = tn << 4;
  int m0 = blockIdx.y << 6;            // 64 rows per wave

  int lane = threadIdx.x & 31;
  int lh   = lane >> 4;                // K offset {0,2} / output rows {0..7, 8..15}
  int idx  = lane & 15;                // M index (A frag), N index (B frag)

  const v8f zero8 = {0.f, 0.f, 0.f, 0.f, 0.f, 0.f, 0.f, 0.f};
  v8f acc[4];
#pragma unroll
  for (int t = 0; t < 4; ++t) acc[t] = zero8;

  const float* arow = A + (long long)(m0 + idx) * lda;
  const long long astep = (long long)16 * lda;

  for (int kb = 0; kb < K; kb += 4) {
    int k0 = kb + lh * 2;
    v2f b;
    if (TRANS_B) {
      const float* bp = B + (long long)(n0 + idx) * ldb + k0;
      b.x = bp[0];
      b.y = bp[1];
    } else {
      const float* bp = B + (long long)k0 * ldb + (n0 + idx);
      b.x = bp[0];
      b.y = bp[ldb];
    }
#pragma unroll
    for (int t = 0; t < 4; ++t) {
      const float* ap = arow + t * astep + k0;
      v2f a;
      a.x = ap[0];
      a.y = ap[1];
      acc[t] = wmma_f32(a, b, acc[t]);
    }
  }

  float bv = bias ? bias[n0 + idx] : 0.f;
#pragma unroll
  for (int t = 0; t < 4; ++t) {
#pragma unroll
    for (int r = 0; r < 8; ++r) {
      int m = m0 + t * 16 + lh * 8 + r;
      float val = acc[t][r] * alpha + bv;
      float* cp = C + (long long)m * ldc + (n0 + idx);
      if (ACCUM) val += *cp;
      *cp = val;
    }
  }
}

// ---------------- repack point projections ----------------
__global__ void repack_pts(const float* __restrict__ qpraw,
                           const float* __restrict__ kvpraw,
                           const float* __restrict__ trans,
                           float* __restrict__ qpts, float* __restrict__ kpts,
                           float* __restrict__ vpts)
{
  int l = blockIdx.x;
  int t = threadIdx.x; // 384 threads
  float t3[3] = {trans[l * 3 + 0], trans[l * 3 + 1], trans[l * 3 + 2]};

  if (t < NH * NPQ * 3) { // 144
    int h = t / 12, rem = t % 12, p = rem / 3, ax = rem % 3;
    qpts[(long long)l * 144 + t] =
        qpraw[(long long)l * 144 + ax * (NH * NPQ) + h * NPQ + p] + t3[ax];
    kpts[(long long)l * 144 + t] =
        kvpraw[(long long)l * 432 + ax * (NH * 12) + h * 12 + p] + t3[ax];
  }
  { // v_pts: per-head padded to 32 columns -> (L, 384)
    int h = t >> 5, c = t & 31;
    float v = 0.f;
    if (c < NPV * 3) {
      int p = c / 3, ax = c % 3;
      v = kvpraw[(long long)l * 432 + ax * (NH * 12) + h * 12 + 4 + p] + t3[ax];
    }
    vpts[(long long)l * DDIM + t] = v;
  }
}

// ---------------- fused pair-bias + point attention + mask (first z stream) ----------------
// block(256) handles (i, 64 j's). z tile (16KB) and k_pts tile (36KB) are
// contiguous in memory -> staged to LDS via the Tensor Data Mover (one DMA each,
// issued by wave 0, completion via s_wait_tensorcnt then workgroup barrier).
__global__ void bias_ptatt(const float* __restrict__ z, const float* __restrict__ wb,
                           const float* __restrict__ bb, const float* __restrict__ hwraw,
                           const float* __restrict__ qpts, const float* __restrict__ kpts,
                           const unsigned char* __restrict__ mask,
                           float* __restrict__ attn)
{
  __shared__ __align__(16) float zt[64 * 64];
  __shared__ __align__(16) float kp[64 * 144];
  __shared__ float qp[144];
  __shared__ float wbs[NH * CZD];
  __shared__ float hws[NH];
  __shared__ float bbs[NH];

  int i  = blockIdx.y;
  int j0 = blockIdx.x * 64;
  int t  = threadIdx.x;

  const float* zsrc = z + ((long long)i * LSEQ + j0) * CZD;      // 4096 dwords, contiguous
  const float* ksrc = kpts + (long long)j0 * 144;                // 9216 dwords, contiguous

#if __has_builtin(__builtin_amdgcn_tensor_load_to_lds)
  if (t < 32) {                 // wave 0 issues both TDM descriptors
    tdm_load_1d(zsrc, zt, 64 * 64);
    tdm_load_1d(ksrc, kp, 64 * 144);
    __builtin_amdgcn_s_wait_tensorcnt(0);
  }
#else
  for (int e = t; e < 64 * 64; e += 256) zt[e] = zsrc[e];
  for (int e = t; e < 64 * 144; e += 256) kp[e] = ksrc[e];
#endif

  for (int e = t; e < 144; e += 256) qp[e] = qpts[(long long)i * 144 + e];
  for (int e = t; e < NH * CZD; e += 256) wbs[e] = wb[e];
  if (t < NH) {
    float w  = hwraw[t];
    float sp = (w > 20.f) ? w : log1pf(__expf(w)); // softplus
    hws[t]   = sp * sqrtf(1.f / (3.f * (NPQ * 9.f / 2.f)));
    bbs[t]   = bb[t];
  }
  __syncthreads();

  for (int o = t; o < 64 * NH; o += 256) {
    int h = o / 64, jl = o % 64;
    int j = j0 + jl;
    float acc = bbs[h];
    const float* zr = &zt[jl * 64];
    const float* wr = &wbs[h * 64];
#pragma unroll 8
    for (int c = 0; c < CZD; ++c) acc += zr[c] * wr[c];

    float d2 = 0.f;
    const float* kr = &kp[jl * 144 + h * 12];
    const float* qr = &qp[h * 12];
#pragma unroll
    for (int e = 0; e < 12; ++e) { float d = qr[e] - kr[e]; d2 += d * d; }
    acc += -0.5f * hws[h] * d2;

    if (!mask[j]) acc += -__builtin_inff();
    attn[(long long)h * LSEQ * LSEQ + (long long)i * LSEQ + j] += acc;
  }
}

// ---------------- row softmax over 768 logits ----------------
__global__ void softmax_rows(float* __restrict__ attn)
{
  long long row = blockIdx.x;
  float* p = attn + row * LSEQ;
  int t = threadIdx.x; // 256
  float a0 = p[t], a1 = p[t + 256], a2 = p[t + 512];

  float mx = fmaxf(a0, fmaxf(a1, a2));
  for (int o = 16; o; o >>= 1) mx = fmaxf(mx, __shfl_xor(mx, o, 32));
  __shared__ float smax[8];
  if ((t & 31) == 0) smax[t >> 5] = mx;
  __syncthreads();
  float m2 = smax[0];
#pragma unroll
  for (int w = 1; w < 8; ++w) m2 = fmaxf(m2, smax[w]);

  a0 = __expf(a0 - m2); a1 = __expf(a1 - m2); a2 = __expf(a2 - m2);
  float s = a0 + a1 + a2;
  for (int o = 16; o; o >>= 1) s += __shfl_xor(s, o, 32);
  __shared__ float ssum[8];
  __syncthreads();
  if ((t & 31) == 0) ssum[t >> 5] = s;
  __syncthreads();
  float st = 0.f;
#pragma unroll
  for (int w = 0; w < 8; ++w) st += ssum[w];
  float inv = 1.f / st;
  p[t] = a0 * inv; p[t + 256] = a1 * inv; p[t + 512] = a2 * inv;
}

// ---------------- z_out (second z stream, on the matrix pipe) ----------------
// per query i:  C(16x64) = A(16x768) * B(768x64), rows m = head (12 valid),
// A[m][k] = attn[m,i,k], B[k][n] = z[i,k,n].  4 waves/block, one 16-col tile each.
// Invalid head rows: clamp the address (stay in-bounds), zero via v_cndmask.
__global__ void zout_wmma(const float* __restrict__ attn, const float* __restrict__ z,
                          float* __restrict__ zout)
{
  int i    = blockIdx.x;
  int wave = threadIdx.x >> 5;
  int n0   = wave << 4; // 0,16,32,48
  int lane = threadIdx.x & 31;
  int lh   = lane >> 4, idx = lane & 15;

  const v8f zero8 = {0.f, 0.f, 0.f, 0.f, 0.f, 0.f, 0.f, 0.f};
  v8f acc = zero8;
  bool mvalid = (idx < NH);
  int hcl = mvalid ? idx : (NH - 1);
  const float* arow = attn + (long long)hcl * LSEQ * LSEQ + (long long)i * LSEQ;

  for (int kb = 0; kb < LSEQ; kb += 4) {
    int k0 = kb + lh * 2;
    float ax = arow[k0];
    float ay = arow[k0 + 1];
    v2f a, b;
    a.x = mvalid ? ax : 0.f;    // v_cndmask, no exec branching
    a.y = mvalid ? ay : 0.f;
    const float* bp = z + ((long long)i * LSEQ + k0) * CZD + n0 + idx;
    b.x = bp[0];
    b.y = bp[CZD];
    acc = wmma_f32(a, b, acc);
  }
#pragma unroll
  for (int r = 0; r < 8; ++r) {
    int m = lh * 8 + r;
    if (m < NH)
      zout[(long long)i * (NH * CZD) + m * CZD + n0 + idx] = acc[r];
  }
}

// ---------------- pt_feat assembly: subtract trans, norm, concat ----------------
__global__ void ptfeat_kernel(const float* __restrict__ optp, const float* __restrict__ trans,
                              float* __restrict__ ptfeat)
{
  int i = blockIdx.x, c = threadIdx.x; // 96 threads: c = h*8+p
  int h = c >> 3, p = c & 7;
  const float* b = optp + (long long)i * DDIM + h * 32 + p * 3;
  float tx = trans[i * 3 + 0], ty = trans[i * 3 + 1], tz = trans[i * 3 + 2];
  float ox = b[0] - tx, oy = b[1] - ty, oz = b[2] - tz;
  float nr = sqrtf(ox * ox + oy * oy + oz * oz + 1e-6f);
  float* o = ptfeat + (long long)i * DDIM;
  o[c] = ox; o[c + 96] = oy; o[c + 192] = oz; o[c + 288] = nr;
}

// ---------------- host-side GEMM launcher ----------------
static void launch_gemm(bool transB, bool accum,
                        const float* A, int lda, long long sA,
                        const float* B, int ldb, long long sB,
                        float* C, int ldc, long long sC,
                        const float* bias, int M, int N, int K, float alpha,
                        int batch, hipStream_t s)
{
  dim3 grid(((N / 16) + 3) / 4, M / 64, batch);  // M is a multiple of 64 in all uses
  dim3 block(128);
  if (transB) {
    if (accum) gemm_wmma<true, true><<<grid, block, 0, s>>>(A, lda, sA, B, ldb, sB, C, ldc, sC, bias, M, N, K, alpha);
    else       gemm_wmma<true, false><<<grid, block, 0, s>>>(A, lda, sA, B, ldb, sB, C, ldc, sC, bias, M, N, K, alpha);
  } else {
    if (accum) gemm_wmma<false, true><<<grid, block, 0, s>>>(A, lda, sA, B, ldb, sB, C, ldc, sC, bias, M, N, K, alpha);
    else       gemm_wmma<false, false><<<grid, block, 0, s>>>(A, lda, sA, B, ldb, sB, C, ldc, sC, bias, M, N, K, alpha);
  }
}

extern "C" void kernel_launch(void* const* d_in, const int* in_sizes, int n_in,
                              void* d_out, int out_size, void* d_ws, size_t ws_size,
                              hipStream_t stream)
{
  const float* x      = (const float*)d_in[0];
  const float* z      = (const float*)d_in[1];
  const float* trans  = (const float*)d_in[2];
  const unsigned char* mask = (const unsigned char*)d_in[3];
  const float* w_q    = (const float*)d_in[4];
  const float* w_k    = (const float*)d_in[5];
  const float* w_v    = (const float*)d_in[6];
  const float* w_o    = (const float*)d_in[7];
  const float* w_b    = (const float*)d_in[8];
  const float* b_b    = (const float*)d_in[9];
  const float* w_z    = (const float*)d_in[10];
  const float* b_z    = (const float*)d_in[11];
  const float* w_qp   = (const float*)d_in[12];
  const float* b_qp   = (const float*)d_in[13];
  const float* w_kvp  = (const float*)d_in[14];
  const float* b_kvp  = (const float*)d_in[15];
  const float* w_pt   = (const float*)d_in[16];
  const float* b_pt   = (const float*)d_in[17];
  const float* hwts   = (const float*)d_in[18];
  float* out = (float*)d_out;

  // workspace layout (floats); total ~10.4M floats (~42 MB)
  float* ws     = (float*)d_ws;
  float* qb     = ws;  ws += (long long)LSEQ * DDIM;
  float* kb     = ws;  ws += (long long)LSEQ * DDIM;
  float* vb     = ws;  ws += (long long)LSEQ * DDIM;
  float* qpraw  = ws;  ws += (long long)LSEQ * 144;
  float* kvpraw = ws;  ws += (long long)LSEQ * 432;
  float* qpts   = ws;  ws += (long long)LSEQ * 144;
  float* kpts   = ws;  ws += (long long)LSEQ * 144;
  float* vpts   = ws;  ws += (long long)LSEQ * DDIM;
  float* attn   = ws;  ws += (long long)NH * LSEQ * LSEQ;
  float* outb   = ws;  ws += (long long)LSEQ * DDIM;
  float* optp   = ws;  ws += (long long)LSEQ * DDIM;
  float* zoutb  = ws;  ws += (long long)LSEQ * NH * CZD;
  float* ptfeat = ws;  ws += (long long)LSEQ * DDIM;

  const long long LL = (long long)LSEQ * LSEQ;

  // 1) input projections  C = x @ W^T (+bias)
  launch_gemm(true, false, x, DDIM, 0, w_q,   DDIM, 0, qb,     DDIM, 0, nullptr, LSEQ, DDIM, DDIM, 1.f, 1, stream);
  launch_gemm(true, false, x, DDIM, 0, w_k,   DDIM, 0, kb,     DDIM, 0, nullptr, LSEQ, DDIM, DDIM, 1.f, 1, stream);
  launch_gemm(true, false, x, DDIM, 0, w_v,   DDIM, 0, vb,     DDIM, 0, nullptr, LSEQ, DDIM, DDIM, 1.f, 1, stream);
  launch_gemm(true, false, x, DDIM, 0, w_qp,  DDIM, 0, qpraw,  144,  0, b_qp,    LSEQ, 144,  DDIM, 1.f, 1, stream);
  launch_gemm(true, false, x, DDIM, 0, w_kvp, DDIM, 0, kvpraw, 432,  0, b_kvp,   LSEQ, 432,  DDIM, 1.f, 1, stream);

  // 2) split/pad point projections (+trans)
  repack_pts<<<LSEQ, 384, 0, stream>>>(qpraw, kvpraw, trans, qpts, kpts, vpts);

  // 3) per-head QK^T logits, scaled by 1/sqrt(D)
  launch_gemm(true, false, qb, DDIM, DH, kb, DDIM, DH, attn, LSEQ, LL, nullptr,
              LSEQ, LSEQ, DH, 1.0f / sqrtf((float)DDIM), NH, stream);

  // 4) + pair bias (z@w_b^T+b_b) + point attention + mask   (first z stream, TDM-staged)
  bias_ptatt<<<dim3(LSEQ / 64, LSEQ), 256, 0, stream>>>(z, w_b, b_b, hwts, qpts, kpts, mask, attn);

  // 5) softmax over j
  softmax_rows<<<NH * LSEQ, 256, 0, stream>>>(attn);

  // 6) out = attn @ v ; o_pt = attn @ v_pts  (per head, N=32)
  launch_gemm(false, false, attn, LSEQ, LL, vb,   DDIM, DH, outb, DDIM, DH, nullptr, LSEQ, DH, LSEQ, 1.f, NH, stream);
  launch_gemm(false, false, attn, LSEQ, LL, vpts, DDIM, DH, optp, DDIM, DH, nullptr, LSEQ, DH, LSEQ, 1.f, NH, stream);

  // 7) z_out = attn-weighted z  (second z stream, WMMA)
  zout_wmma<<<LSEQ, 128, 0, stream>>>(attn, z, zoutb);

  // 8) pt_feat assembly
  ptfeat_kernel<<<LSEQ, 96, 0, stream>>>(optp, trans, ptfeat);

  // 9) final = out@w_o^T + (z_out@w_z^T + b_z) + (pt_feat@w_pt^T + b_pt)
  launch_gemm(true, false, outb,   DDIM,     0, w_o,  DDIM,     0, out, DDIM, 0, nullptr, LSEQ, DDIM, DDIM,     1.f, 1, stream);
  launch_gemm(true, true,  zoutb,  NH * CZD, 0, w_z,  NH * CZD, 0, out, DDIM, 0, b_z,     LSEQ, DDIM, NH * CZD, 1.f, 1, stream);
  launch_gemm(true, true,  ptfeat, DDIM,     0, w_pt, DDIM,     0, out, DDIM, 0, b_pt,    LSEQ, DDIM, DDIM,     1.f, 1, stream);
}